// _PlainBlockGAT_59906203845000
// MI455X (gfx1250) — compile-verified
//
#include <hip/hip_runtime.h>
#include <hip/hip_bf16.h>
#include <math.h>

#define NN   50000
#define EE   800000
#define DIN  128
#define DOUTC 128
#define HH   4
#define CCH  32
#define GG   64
#define ETOT (EE + NN)

typedef __attribute__((ext_vector_type(2))) float v2f;
typedef __attribute__((ext_vector_type(8))) float v8f;

__device__ __forceinline__ int f2ord(float f) {
    int i = __float_as_int(f);
    return i < 0 ? (i ^ 0x7FFFFFFF) : i;
}
__device__ __forceinline__ float ord2f(int i) {
    return __int_as_float(i < 0 ? (i ^ 0x7FFFFFFF) : i);
}

// ---------------------------------------------------------------- init
__global__ void k_init(float* __restrict__ out, float* __restrict__ denom,
                       int* __restrict__ amax, float* __restrict__ gsum,
                       float* __restrict__ vsum, float* __restrict__ cnt) {
    int idx = blockIdx.x * blockDim.x + threadIdx.x;
    if (idx < NN * DOUTC) out[idx] = 0.0f;
    if (idx < NN * HH) { denom[idx] = 0.0f; amax[idx] = (int)0x80000000; }
    if (idx < GG * DOUTC) { gsum[idx] = 0.0f; vsum[idx] = 0.0f; }
    if (idx < GG) cnt[idx] = 0.0f;
}

// ------------------------------------------------- dual GEMM via WMMA f32
// xl = x @ Wl, xr = x @ Wr.  One wave per 16x16 output tile, K stepped by 4.
// A 16x4 f32: lanes 0-15 -> K={k0,k0+1}, lanes 16-31 -> K={k0+2,k0+3}
// B 4x16 f32: vgpr j, lane L -> K = k0 + j + (L/16)*2, N = n0 + L%16
// C 16x16 f32: vgpr v, lane L -> M = m0 + v + (L/16)*8, N = n0 + L%16
__global__ void k_gemm_wmma(const float* __restrict__ x,
                            const float* __restrict__ Wl,
                            const float* __restrict__ Wr,
                            float* __restrict__ xl, float* __restrict__ xr) {
    const int ROW_TILES = NN / 16;            // 3125 (exact)
    const int COL_TILES = DOUTC / 16;         // 8
    const int TILES     = ROW_TILES * COL_TILES;  // 25000 per matrix
    int wave = (blockIdx.x * blockDim.x + threadIdx.x) >> 5;
    int lane = threadIdx.x & 31;
    if (wave >= 2 * TILES) return;            // wave-uniform (exact anyway)
    const float* W = (wave < TILES) ? Wl : Wr;
    float*       O = (wave < TILES) ? xl : xr;
    int t   = wave % TILES;
    int m0  = (t / COL_TILES) * 16;
    int n0  = (t % COL_TILES) * 16;
    int row = lane & 15;
    int khi = (lane >> 4) << 1;               // 0 or 2

    v8f c = {0.f, 0.f, 0.f, 0.f, 0.f, 0.f, 0.f, 0.f};
    #pragma unroll
    for (int k0 = 0; k0 < DIN; k0 += 4) {
        int ka = k0 + khi;
        v2f a, b;
        a.x = x[(size_t)(m0 + row) * DIN + ka];
        a.y = x[(size_t)(m0 + row) * DIN + ka + 1];
        b.x = W[(size_t)(ka + 0) * DOUTC + n0 + row];
        b.y = W[(size_t)(ka + 1) * DOUTC + n0 + row];
        c = __builtin_amdgcn_wmma_f32_16x16x4_f32(false, a, false, b,
                                                  (short)0, c, false, false);
    }
    int mhalf = (lane >> 4) << 3;             // 0 or 8
    #pragma unroll
    for (int v = 0; v < 8; ++v)
        O[(size_t)(m0 + v + mhalf) * DOUTC + n0 + row] = c[v];
}

// --------------------------------- per-edge logits + segment max (1 wave/edge)
__global__ void k_edge_logits(const float* __restrict__ xl,
                              const float* __restrict__ xr,
                              const int* __restrict__ ei,
                              const float* __restrict__ att,
                              float* __restrict__ logits,
                              int* __restrict__ amax) {
    int wave = (blockIdx.x * blockDim.x + threadIdx.x) >> 5;
    int lane = threadIdx.x & 31;
    if (wave >= ETOT) return;
    int s, d;
    if (wave < EE) { s = ei[wave]; d = ei[EE + wave]; }
    else           { s = d = wave - EE; }

    const float4 a = *(const float4*)(xl + (size_t)s * DOUTC + lane * 4);
    const float4 b = *(const float4*)(xr + (size_t)d * DOUTC + lane * 4);
    const float4 w = *(const float4*)(att + lane * 4);
    float m, e, sum = 0.0f;
    m = a.x + b.x; e = (m > 0.0f) ? m : 0.2f * m; sum += e * w.x;
    m = a.y + b.y; e = (m > 0.0f) ? m : 0.2f * m; sum += e * w.y;
    m = a.z + b.z; e = (m > 0.0f) ? m : 0.2f * m; sum += e * w.z;
    m = a.w + b.w; e = (m > 0.0f) ? m : 0.2f * m; sum += e * w.w;
    // reduce the 8 lanes of each head (lanes h*8 .. h*8+7 cover head h)
    sum += __shfl_xor(sum, 1);
    sum += __shfl_xor(sum, 2);
    sum += __shfl_xor(sum, 4);
    if ((lane & 7) == 0) {
        int h = lane >> 3;
        logits[(size_t)wave * HH + h] = sum;
        atomicMax(&amax[d * HH + h], f2ord(sum));
    }
}

// --------------------------------- exp(logit - max) + segment denom
__global__ void k_edge_exp(float* __restrict__ logits,   // in: logits, out: ex
                           const int* __restrict__ ei,
                           const int* __restrict__ amax,
                           float* __restrict__ denom) {
    int idx = blockIdx.x * blockDim.x + threadIdx.x;
    if (idx >= ETOT * HH) return;
    int e = idx >> 2, h = idx & 3;
    int d = (e < EE) ? ei[EE + e] : (e - EE);
    float mx = ord2f(amax[d * HH + h]);
    float ex = expf(logits[idx] - mx);
    logits[idx] = ex;
    atomicAdd(&denom[d * HH + h], ex);
}

// --------------------------------- alpha-weighted scatter add (1 wave/edge)
__global__ void k_edge_scatter(const float* __restrict__ xl,
                               const float* __restrict__ ex,
                               const float* __restrict__ denom,
                               const int* __restrict__ ei,
                               float* __restrict__ out) {
    int wave = (blockIdx.x * blockDim.x + threadIdx.x) >> 5;
    int lane = threadIdx.x & 31;
    if (wave >= ETOT) return;
    int s, d;
    if (wave < EE) { s = ei[wave]; d = ei[EE + wave]; }
    else           { s = d = wave - EE; }
    int h = lane >> 3;
    float alpha = ex[(size_t)wave * HH + h] / (denom[d * HH + h] + 1e-16f);
    const float4 a = *(const float4*)(xl + (size_t)s * DOUTC + lane * 4);
    float* o = out + (size_t)d * DOUTC + lane * 4;
    atomicAdd(o + 0, alpha * a.x);
    atomicAdd(o + 1, alpha * a.y);
    atomicAdd(o + 2, alpha * a.z);
    atomicAdd(o + 3, alpha * a.w);
}

// --------------------------------- GraphNorm: per-graph sums + counts
__global__ void k_stats_sum(const float* __restrict__ out,
                            const float* __restrict__ bias,
                            const int* __restrict__ batch,
                            float* __restrict__ gsum, float* __restrict__ cnt) {
    int idx = blockIdx.x * blockDim.x + threadIdx.x;
    if (idx >= NN * DOUTC) return;
    int n = idx >> 7, j = idx & 127;
    int g = batch[n];
    float v = out[idx] + bias[j];
    atomicAdd(&gsum[g * DOUTC + j], v);
    if (j == 0) atomicAdd(&cnt[g], 1.0f);
}

__global__ void k_mean(float* __restrict__ gsum, const float* __restrict__ cnt) {
    int idx = blockIdx.x * blockDim.x + threadIdx.x;
    if (idx >= GG * DOUTC) return;
    gsum[idx] = gsum[idx] / fmaxf(cnt[idx >> 7], 1.0f);   // gsum -> mean
}

__global__ void k_var(const float* __restrict__ out,
                      const float* __restrict__ bias,
                      const int* __restrict__ batch,
                      const float* __restrict__ mean,
                      const float* __restrict__ mscale,
                      float* __restrict__ vsum) {
    int idx = blockIdx.x * blockDim.x + threadIdx.x;
    if (idx >= NN * DOUTC) return;
    int n = idx >> 7, j = idx & 127;
    int g = batch[n];
    float v  = out[idx] + bias[j];
    float dc = v - mean[g * DOUTC + j] * mscale[j];
    atomicAdd(&vsum[g * DOUTC + j], dc * dc);
}

__global__ void k_rstd(float* __restrict__ vsum, const float* __restrict__ cnt) {
    int idx = blockIdx.x * blockDim.x + threadIdx.x;
    if (idx >= GG * DOUTC) return;
    float var = vsum[idx] / fmaxf(cnt[idx >> 7], 1.0f);
    vsum[idx] = rsqrtf(var + 1e-5f);                      // vsum -> rstd
}

__global__ void k_final(float* __restrict__ out,          // in: accum, out: y
                        const float* __restrict__ bias,
                        const int* __restrict__ batch,
                        const float* __restrict__ mean,
                        const float* __restrict__ rstd,
                        const float* __restrict__ mscale,
                        const float* __restrict__ gw,
                        const float* __restrict__ gb) {
    int idx = blockIdx.x * blockDim.x + threadIdx.x;
    if (idx >= NN * DOUTC) return;
    int n = idx >> 7, j = idx & 127;
    int g = batch[n];
    float v  = out[idx] + bias[j];
    float dc = v - mean[g * DOUTC + j] * mscale[j];
    float y  = dc * rstd[g * DOUTC + j] * gw[j] + gb[j];
    out[idx] = (y > 0.0f) ? y : expm1f(y);                // ELU
}

// ----------------------------------------------------------------- launch
extern "C" void kernel_launch(void* const* d_in, const int* in_sizes, int n_in,
                              void* d_out, int out_size, void* d_ws, size_t ws_size,
                              hipStream_t stream) {
    const float* x     = (const float*)d_in[0];
    const int*   ei    = (const int*)  d_in[1];
    const int*   batch = (const int*)  d_in[2];
    const float* Wl    = (const float*)d_in[3];
    const float* Wr    = (const float*)d_in[4];
    const float* att   = (const float*)d_in[5];
    const float* bias  = (const float*)d_in[6];
    const float* gw    = (const float*)d_in[7];
    const float* gb    = (const float*)d_in[8];
    const float* gms   = (const float*)d_in[9];
    float* out = (float*)d_out;

    char* ws = (char*)d_ws;
    size_t off = 0;
    auto carve = [&](size_t bytes) -> void* {
        void* p = ws + off;
        off = (off + bytes + 255) & ~(size_t)255;
        return p;
    };
    float* xl     = (float*)carve(sizeof(float) * (size_t)NN * DOUTC);   // 25.6 MB
    float* xr     = (float*)carve(sizeof(float) * (size_t)NN * DOUTC);   // 25.6 MB
    float* logits = (float*)carve(sizeof(float) * (size_t)ETOT * HH);    // 13.6 MB
    int*   amax   = (int*)  carve(sizeof(int)   * (size_t)NN * HH);
    float* denom  = (float*)carve(sizeof(float) * (size_t)NN * HH);
    float* gsum   = (float*)carve(sizeof(float) * GG * DOUTC);
    float* vsum   = (float*)carve(sizeof(float) * GG * DOUTC);
    float* cnt    = (float*)carve(sizeof(float) * GG);
    (void)ws_size; (void)in_sizes; (void)n_in; (void)out_size;

    // 1) init accumulators
    k_init<<<(NN * DOUTC + 255) / 256, 256, 0, stream>>>(out, denom, amax, gsum, vsum, cnt);

    // 2) xl = x@Wl, xr = x@Wr  (WMMA f32 16x16x4; 50000 waves, 4 waves/block)
    k_gemm_wmma<<<(2 * (NN / 16) * (DOUTC / 16)) / 4, 128, 0, stream>>>(x, Wl, Wr, xl, xr);

    // 3) edge logits + segment max (1 wave/edge; ETOT divisible by 8)
    k_edge_logits<<<ETOT / 8, 256, 0, stream>>>(xl, xr, ei, att, logits, amax);

    // 4) exp + segment denom
    k_edge_exp<<<(ETOT * HH + 255) / 256, 256, 0, stream>>>(logits, ei, amax, denom);

    // 5) alpha-weighted scatter add into out
    k_edge_scatter<<<ETOT / 8, 256, 0, stream>>>(xl, logits, denom, ei, out);

    // 6) GraphNorm stats
    k_stats_sum<<<(NN * DOUTC + 255) / 256, 256, 0, stream>>>(out, bias, batch, gsum, cnt);
    k_mean<<<(GG * DOUTC + 255) / 256, 256, 0, stream>>>(gsum, cnt);
    k_var<<<(NN * DOUTC + 255) / 256, 256, 0, stream>>>(out, bias, batch, gsum, gms, vsum);
    k_rstd<<<(GG * DOUTC + 255) / 256, 256, 0, stream>>>(vsum, cnt);

    // 7) normalize + affine + ELU (in place on d_out)
    k_final<<<(NN * DOUTC + 255) / 256, 256, 0, stream>>>(out, bias, batch, gsum, vsum, gms, gw, gb);
}